// QuantumLayer_25005299597353
// MI455X (gfx1250) — compile-verified
//
#include <hip/hip_runtime.h>

typedef __attribute__((ext_vector_type(2))) float v2f;
typedef __attribute__((ext_vector_type(8))) float v8f;

#define NW 16
#define NSTATE 65536            // 2^16
#define BATCH 128
#define NLAYERS 6

#define WMMA_F32X4(A, B, C) \
    __builtin_amdgcn_wmma_f32_16x16x4_f32(false, (A), false, (B), (short)0, (C), false, false)

// GF(2) linear map on 16 index bits: rows m[k] = image of basis vector e_k.
struct PermM { unsigned short m[NW]; };

__device__ __forceinline__ unsigned perm_apply(const PermM& M, unsigned j, int nbits) {
    unsigned v = 0;
#pragma unroll
    for (int k = 0; k < NW; ++k)
        if (k < nbits && ((j >> k) & 1u)) v ^= M.m[k];
    return v;
}

// ---------------------------------------------------------------------------
// 1) Init: post-embedding product state  psi[j] = prod_i (bit_i(j) ? s_i : c_i)
// grid (256, 128), block 256
// ---------------------------------------------------------------------------
__global__ void qsim_init_state(const float* __restrict__ x, float2* __restrict__ state) {
    __shared__ float cs[NW], sn[NW];
    const int b = blockIdx.y;
    const int tid = threadIdx.x;
    if (tid < NW) {
        float a = 0.5f * x[b * NW + tid];
        cs[tid] = cosf(a);
        sn[tid] = sinf(a);
    }
    __syncthreads();
    const int j = blockIdx.x * blockDim.x + tid;
    float p = 1.0f;
#pragma unroll
    for (int i = 0; i < NW; ++i) {
        int bit = (j >> (15 - i)) & 1;
        p *= bit ? sn[i] : cs[i];
    }
    state[(size_t)b * NSTATE + j] = float2{p, 0.0f};
}

// ---------------------------------------------------------------------------
// 2) Build fused 4-wire unitaries for one layer: U[g][s][t], g=0..3, 16x16 cmplx
// ---------------------------------------------------------------------------
__global__ void qsim_build_u(const float* __restrict__ wl, float2* __restrict__ U) {
    const int tid = threadIdx.x;
    const int s = tid >> 4;
    const int t = tid & 15;
    for (int g = 0; g < 4; ++g) {
        float pr = 1.0f, pi = 0.0f;
#pragma unroll
        for (int k = 0; k < 4; ++k) {
            const int wire = 4 * g + k;
            const float phi = wl[wire * 3 + 0];
            const float th  = wl[wire * 3 + 1];
            const float om  = wl[wire * 3 + 2];
            const float c  = cosf(0.5f * th);
            const float ss = sinf(0.5f * th);
            const float a  = 0.5f * (phi + om);
            const float bb = 0.5f * (phi - om);
            const int sb = (s >> (3 - k)) & 1;
            const int tb = (t >> (3 - k)) & 1;
            float er, ei;
            if (sb == 0 && tb == 0)      { er =  c  * cosf(a);  ei = -c  * sinf(a);  }
            else if (sb == 0 && tb == 1) { er = -ss * cosf(bb); ei = -ss * sinf(bb); }
            else if (sb == 1 && tb == 0) { er =  ss * cosf(bb); ei = -ss * sinf(bb); }
            else                         { er =  c  * cosf(a);  ei =  c  * sinf(a);  }
            const float nr = pr * er - pi * ei;
            const float ni = pr * ei + pi * er;
            pr = nr; pi = ni;
        }
        U[g * 256 + s * 16 + t] = float2{pr, pi};
    }
}

// ---------------------------------------------------------------------------
// 3a) In-place fused 4-wire unitary: 16x16 complex GEMM, one 16-col tile/wave,
//     16x v_wmma_f32_16x16x4_f32 per tile. p = nibble position of target bits.
//     grid 4096, block 256 (8 waves) -> 32768 waves = 128 batches * 256 tiles
// ---------------------------------------------------------------------------
__global__ void qsim_apply_group(float2* __restrict__ state,
                                 const float2* __restrict__ U, int p) {
    const int lane = threadIdx.x & 31;
    const int wv   = threadIdx.x >> 5;
    const int gw   = blockIdx.x * (blockDim.x >> 5) + wv;
    const int batch = gw >> 8;
    const int tile  = gw & 255;
    const int h = lane >> 4;
    const int n = lane & 15;

    const int m = tile * 16 + n;                 // column 0..4095
    const int shift = 4 * p;
    const int lo = m & ((1 << shift) - 1);
    const int hi = m >> shift;
    const int basej = (hi << (shift + 4)) | lo;

    float2* st = state + (size_t)batch * NSTATE;

    v2f Br[4], Bi[4], Ar[4], Ai[4], An[4];
#pragma unroll
    for (int kc = 0; kc < 4; ++kc) {
        const int t0 = 4 * kc + 2 * h;
        float2 e0 = st[basej + (t0 << shift)];
        float2 e1 = st[basej + ((t0 + 1) << shift)];
        Br[kc] = v2f{e0.x, e1.x};
        Bi[kc] = v2f{e0.y, e1.y};
        float2 u0 = U[n * 16 + t0];
        float2 u1 = U[n * 16 + t0 + 1];
        Ar[kc] = v2f{u0.x, u1.x};
        Ai[kc] = v2f{u0.y, u1.y};
        An[kc] = v2f{-u0.y, -u1.y};
    }

    v8f Dr = {};
    v8f Di = {};
#pragma unroll
    for (int kc = 0; kc < 4; ++kc) Dr = WMMA_F32X4(Ar[kc], Br[kc], Dr);
#pragma unroll
    for (int kc = 0; kc < 4; ++kc) Dr = WMMA_F32X4(An[kc], Bi[kc], Dr);
#pragma unroll
    for (int kc = 0; kc < 4; ++kc) Di = WMMA_F32X4(Ar[kc], Bi[kc], Di);
#pragma unroll
    for (int kc = 0; kc < 4; ++kc) Di = WMMA_F32X4(Ai[kc], Br[kc], Di);

#pragma unroll
    for (int v = 0; v < 8; ++v) {
        const int s = v + 8 * h;
        st[basej + (s << shift)] = float2{Dr[v], Di[v]};
    }
}

// ---------------------------------------------------------------------------
// 3b) Gather variant for group 0 (p=3, top nibble): additionally applies the
//     PREVIOUS layer's CNOT-ring permutation by reading src[Finv(j)], writing
//     dst[j]. Minv = inverse ring permutation as a GF(2) matrix.
// ---------------------------------------------------------------------------
__global__ void qsim_apply_group0_perm(const float2* __restrict__ src,
                                       float2* __restrict__ dst,
                                       const float2* __restrict__ U,
                                       PermM Minv) {
    const int lane = threadIdx.x & 31;
    const int wv   = threadIdx.x >> 5;
    const int gw   = blockIdx.x * (blockDim.x >> 5) + wv;
    const int batch = gw >> 8;
    const int tile  = gw & 255;
    const int h = lane >> 4;
    const int n = lane & 15;

    const int col = tile * 16 + n;               // j = col | (t<<12), col<4096
    const float2* ss = src + (size_t)batch * NSTATE;
    float2* sd = dst + (size_t)batch * NSTATE;

    // Finv(col | t<<12) = Finv(col) ^ Finv(t<<12); precompute pieces.
    const unsigned fb = perm_apply(Minv, (unsigned)col, 12);
    const unsigned i0 = Minv.m[12], i1 = Minv.m[13], i2 = Minv.m[14], i3 = Minv.m[15];

    v2f Br[4], Bi[4], Ar[4], Ai[4], An[4];
#pragma unroll
    for (int kc = 0; kc < 4; ++kc) {
        const int t0 = 4 * kc + 2 * h;
        unsigned n0 = 0, n1 = 0;
        if (t0 & 1) n0 ^= i0;  if (t0 & 2) n0 ^= i1;
        if (t0 & 4) n0 ^= i2;  if (t0 & 8) n0 ^= i3;
        const int t1 = t0 + 1;
        if (t1 & 1) n1 ^= i0;  if (t1 & 2) n1 ^= i1;
        if (t1 & 4) n1 ^= i2;  if (t1 & 8) n1 ^= i3;
        float2 e0 = ss[fb ^ n0];
        float2 e1 = ss[fb ^ n1];
        Br[kc] = v2f{e0.x, e1.x};
        Bi[kc] = v2f{e0.y, e1.y};
        float2 u0 = U[n * 16 + t0];
        float2 u1 = U[n * 16 + t1];
        Ar[kc] = v2f{u0.x, u1.x};
        Ai[kc] = v2f{u0.y, u1.y};
        An[kc] = v2f{-u0.y, -u1.y};
    }

    v8f Dr = {};
    v8f Di = {};
#pragma unroll
    for (int kc = 0; kc < 4; ++kc) Dr = WMMA_F32X4(Ar[kc], Br[kc], Dr);
#pragma unroll
    for (int kc = 0; kc < 4; ++kc) Dr = WMMA_F32X4(An[kc], Bi[kc], Dr);
#pragma unroll
    for (int kc = 0; kc < 4; ++kc) Di = WMMA_F32X4(Ar[kc], Bi[kc], Di);
#pragma unroll
    for (int kc = 0; kc < 4; ++kc) Di = WMMA_F32X4(Ai[kc], Br[kc], Di);

#pragma unroll
    for (int v = 0; v < 8; ++v) {
        const int s = v + 8 * h;
        sd[col | (s << 12)] = float2{Dr[v], Di[v]};
    }
}

// ---------------------------------------------------------------------------
// 4) Fallback in-place CNOT (only used when workspace can't hold two buffers).
// ---------------------------------------------------------------------------
__global__ void qsim_cnot(float2* __restrict__ state, int bc, int bt) {
    const int gid = blockIdx.x * blockDim.x + threadIdx.x;
    const int b = gid >> 14;
    int x = gid & 16383;
    const int p1 = bc < bt ? bc : bt;
    const int p2 = bc < bt ? bt : bc;
    x = ((x >> p1) << (p1 + 1)) | (x & ((1 << p1) - 1));
    x = ((x >> p2) << (p2 + 1)) | (x & ((1 << p2) - 1));
    const int j0 = x | (1 << bc);
    const int j1 = j0 | (1 << bt);
    float2* st = state + (size_t)b * NSTATE;
    float2 a = st[j0];
    float2 c = st[j1];
    st[j0] = c;
    st[j1] = a;
}

// ---------------------------------------------------------------------------
// 5) <Z_i>: stage 1 with deterministic LDS tree. Pending final permutation F
//    applied as a bit-label remap: amplitude at j has basis label F(j).
//    grid (16, 128), block 256; each thread covers 16 amplitudes.
// ---------------------------------------------------------------------------
__global__ void qsim_expz_partial(const float2* __restrict__ state,
                                  float* __restrict__ partial, PermM Mf) {
    __shared__ float red[256 * NW];   // 16 KB
    const int b = blockIdx.y;
    const int chunk = blockIdx.x;
    const int tid = threadIdx.x;
    const float2* st = state + (size_t)b * NSTATE;

    float z[NW];
#pragma unroll
    for (int i = 0; i < NW; ++i) z[i] = 0.0f;

#pragma unroll
    for (int e = 0; e < 16; ++e) {
        const int j = chunk * 4096 + e * 256 + tid;
        float2 a = st[j];
        float pr = a.x * a.x + a.y * a.y;
        const unsigned label = perm_apply(Mf, (unsigned)j, 16);
#pragma unroll
        for (int i = 0; i < NW; ++i)
            z[i] += ((label >> (15 - i)) & 1u) ? -pr : pr;
    }
#pragma unroll
    for (int i = 0; i < NW; ++i) red[tid * NW + i] = z[i];
    __syncthreads();
    for (int off = 128; off > 0; off >>= 1) {
        if (tid < off) {
#pragma unroll
            for (int i = 0; i < NW; ++i)
                red[tid * NW + i] += red[(tid + off) * NW + i];
        }
        __syncthreads();
    }
    if (tid < NW)
        partial[((size_t)b * 16 + chunk) * NW + tid] = red[tid];
}

__global__ void qsim_expz_final(const float* __restrict__ partial,
                                float* __restrict__ out) {
    const int b = blockIdx.x;
    const int i = threadIdx.x;
    if (i < NW) {
        float s = 0.0f;
#pragma unroll
        for (int c = 0; c < 16; ++c)
            s += partial[((size_t)b * 16 + c) * NW + i];
        out[b * NW + i] = s;
    }
}

// ---------------------------------------------------------------------------
// Host side
// ---------------------------------------------------------------------------
static PermM ring_perm_matrix(int r, bool inverse) {
    // Ring of CNOTs (wires): for i=0..15: target (i+r)%16 ^= control i.
    // Index-bit positions: bc = 15-i, bt = 15-((i+r)%16). Inverse = reverse order.
    PermM M;
    for (int k = 0; k < NW; ++k) {
        unsigned v = 1u << k;
        if (!inverse) {
            for (int i = 0; i < NW; ++i) {
                int bc = 15 - i, bt = 15 - ((i + r) % NW);
                v ^= ((v >> bc) & 1u) << bt;
            }
        } else {
            for (int i = NW - 1; i >= 0; --i) {
                int bc = 15 - i, bt = 15 - ((i + r) % NW);
                v ^= ((v >> bc) & 1u) << bt;
            }
        }
        M.m[k] = (unsigned short)v;
    }
    return M;
}

static PermM identity_perm() {
    PermM M;
    for (int k = 0; k < NW; ++k) M.m[k] = (unsigned short)(1u << k);
    return M;
}

extern "C" void kernel_launch(void* const* d_in, const int* in_sizes, int n_in,
                              void* d_out, int out_size, void* d_ws, size_t ws_size,
                              hipStream_t stream) {
    const float* x = (const float*)d_in[0];   // (128, 16)
    const float* w = (const float*)d_in[1];   // (6, 16, 3)
    float* out = (float*)d_out;               // (128, 16)

    const size_t STATE_BYTES = (size_t)BATCH * NSTATE * sizeof(float2);  // 64 MB
    const size_t U_BYTES = 4 * 256 * sizeof(float2);                     // 8 KB
    const size_t PART_BYTES = (size_t)BATCH * 16 * NW * sizeof(float);   // 128 KB
    const bool fused = ws_size >= 2 * STATE_BYTES + U_BYTES + PART_BYTES;

    char* ws = (char*)d_ws;
    float2* buf0 = (float2*)ws;
    float2* buf1 = fused ? (float2*)(ws + STATE_BYTES) : nullptr;
    char* tail = ws + (fused ? 2 * STATE_BYTES : STATE_BYTES);
    float2* U = (float2*)tail;
    float* partial = (float*)(tail + U_BYTES);

    // Embedding as direct product-state init (replaces 16 gate passes).
    qsim_init_state<<<dim3(NSTATE / 256, BATCH), 256, 0, stream>>>(x, buf0);

    if (fused) {
        float2* cur = buf0;
        for (int l = 0; l < NLAYERS; ++l) {
            qsim_build_u<<<1, 256, 0, stream>>>(w + (size_t)l * NW * 3, U);
            if (l == 0) {
                qsim_apply_group<<<4096, 256, 0, stream>>>(cur, U, 3);
            } else {
                // Fold previous layer's CNOT ring into this gather pass.
                const int rprev = ((l - 1) % (NW - 1)) + 1;
                float2* nxt = (cur == buf0) ? buf1 : buf0;
                qsim_apply_group0_perm<<<4096, 256, 0, stream>>>(
                    cur, nxt, U, ring_perm_matrix(rprev, /*inverse=*/true));
                cur = nxt;
            }
            for (int g = 1; g < 4; ++g)
                qsim_apply_group<<<4096, 256, 0, stream>>>(cur, U + g * 256, 3 - g);
        }
        // Layer 5's ring is still pending: fold into readout as a label remap.
        const int rlast = ((NLAYERS - 1) % (NW - 1)) + 1;
        qsim_expz_partial<<<dim3(16, BATCH), 256, 0, stream>>>(
            cur, partial, ring_perm_matrix(rlast, /*inverse=*/false));
    } else {
        // Fallback: single state buffer, physical in-place CNOT passes.
        for (int l = 0; l < NLAYERS; ++l) {
            qsim_build_u<<<1, 256, 0, stream>>>(w + (size_t)l * NW * 3, U);
            for (int g = 0; g < 4; ++g)
                qsim_apply_group<<<4096, 256, 0, stream>>>(buf0, U + g * 256, 3 - g);
            const int r = (l % (NW - 1)) + 1;
            for (int i = 0; i < NW; ++i) {
                const int t = (i + r) % NW;
                qsim_cnot<<<8192, 256, 0, stream>>>(buf0, 15 - i, 15 - t);
            }
        }
        qsim_expz_partial<<<dim3(16, BATCH), 256, 0, stream>>>(
            buf0, partial, identity_perm());
    }
    qsim_expz_final<<<BATCH, 32, 0, stream>>>(partial, out);
}